// EdgeConv_80942953660740
// MI455X (gfx1250) — compile-verified
//
#include <hip/hip_runtime.h>
#include <hip/hip_bf16.h>

typedef __attribute__((ext_vector_type(2))) float v2f;
typedef __attribute__((ext_vector_type(8))) float v8f;

#define NPTS 4096
#define CDIM 64
#define ODIM 64
#define KNN  20
#define BN_EPS 1e-5f
#define NEG_SLOPE 0.2f

#define WMMA_F32(a, b, c) \
    __builtin_amdgcn_wmma_f32_16x16x4_f32(false, (a), false, (b), (short)0, (c), false, false)

// ---------------------------------------------------------------------------
// Kernel A: sq[b][n] = sum_c x[b][c][n]^2
// ---------------------------------------------------------------------------
__global__ __launch_bounds__(256) void sq_kernel(const float* __restrict__ x,
                                                 float* __restrict__ sq) {
    int gid = blockIdx.x * 256 + threadIdx.x;      // gid in [0, B*NPTS)
    int b = gid / NPTS;
    int n = gid % NPTS;
    const float* xb = x + (size_t)b * CDIM * NPTS;
    float s = 0.f;
#pragma unroll
    for (int c = 0; c < CDIM; ++c) {
        float v = xb[c * NPTS + n];
        s += v * v;
    }
    sq[gid] = s;
}

// ---------------------------------------------------------------------------
// Kernel B: KNN.  Block = 512 threads (16 waves) handles 16 query rows of one
// batch; full 16x4096 d2 stripe in 256KB dynamic LDS.
// GEMM: f32 WMMA, two concurrent accumulator chains per wave, B-fragments
// loaded in clauses with one-pair-ahead software pipelining.
// Selection: 20 min-extractions, ds_load_b128 scans + wave32 shfl reduction.
// ---------------------------------------------------------------------------
__global__ __launch_bounds__(512) void knn_kernel(const float* __restrict__ x,
                                                  const float* __restrict__ sq,
                                                  int* __restrict__ idx_out) {
    extern __shared__ float lds[];                 // 16 * 4096 floats
    const int b    = blockIdx.y;
    const int n0   = blockIdx.x * 16;
    const int tid  = threadIdx.x;
    const int wave = tid >> 5;
    const int lane = tid & 31;
    const int half = lane >> 4;                    // 0 or 1
    const int l16  = lane & 15;

    const float* xb  = x  + (size_t)b * CDIM * NPTS;
    const float* sqb = sq + (size_t)b * NPTS;

    // A (query) fragments for all 16 K-steps (K = 64, 4 per step).
    v2f afrag[16];
#pragma unroll
    for (int ks = 0; ks < 16; ++ks) {
        int c0 = ks * 4 + half * 2;
        v2f a;
        a.x = xb[(c0 + 0) * NPTS + n0 + l16];
        a.y = xb[(c0 + 1) * NPTS + n0 + l16];
        afrag[ks] = a;
    }
    float sqn[8];
#pragma unroll
    for (int i = 0; i < 8; ++i) sqn[i] = sqb[n0 + i + 8 * half];

    // Pair p covers column tiles m0a=(wave+32p)*16 and m0b=(wave+16+32p)*16.
    v2f bufA[2][16], bufB[2][16];

#define LOAD_TILE(buf, m0)                                        \
    {                                                             \
        _Pragma("unroll") for (int ks = 0; ks < 16; ++ks) {       \
            int c0 = ks * 4 + half * 2;                           \
            (buf)[ks].x = xb[(c0 + 0) * NPTS + (m0) + l16];       \
            (buf)[ks].y = xb[(c0 + 1) * NPTS + (m0) + l16];       \
        }                                                         \
    }

    LOAD_TILE(bufA[0], wave * 16)
    LOAD_TILE(bufB[0], (wave + 16) * 16)

#pragma unroll
    for (int p = 0; p < 8; ++p) {
        const int cur = p & 1, nxt = cur ^ 1;
        if (p < 7) {                                  // prefetch next pair
            LOAD_TILE(bufA[nxt], (wave + 32 * (p + 1)) * 16)
            LOAD_TILE(bufB[nxt], (wave + 16 + 32 * (p + 1)) * 16)
        }
        const int m0a = (wave + 32 * p) * 16;
        const int m0b = (wave + 16 + 32 * p) * 16;
        v8f acc0 = {}, acc1 = {};
#pragma unroll
        for (int ks = 0; ks < 16; ++ks) {
            acc0 = WMMA_F32(afrag[ks], bufA[cur][ks], acc0);
            acc1 = WMMA_F32(afrag[ks], bufB[cur][ks], acc1);
        }
        float sqma = sqb[m0a + l16];
        float sqmb = sqb[m0b + l16];
#pragma unroll
        for (int i = 0; i < 8; ++i) {
            int row = i + 8 * half;
            lds[row * NPTS + m0a + l16] = sqn[i] + sqma - 2.0f * acc0[i];
            lds[row * NPTS + m0b + l16] = sqn[i] + sqmb - 2.0f * acc1[i];
        }
    }
#undef LOAD_TILE
    __syncthreads();

    // Selection: wave w owns row w.
    {
        float* drow = lds + wave * NPTS;
        const float4* drow4 = reinterpret_cast<const float4*>(drow);
        int* orow = idx_out + ((size_t)b * NPTS + n0 + wave) * KNN;
        for (int kk = 0; kk < KNN; ++kk) {
            float best = 3.4e38f;
            int bidx = 0x7fffffff;
            for (int j = 0; j < 32; ++j) {
                float4 v4 = drow4[lane + 32 * j];
                int mb = 4 * (lane + 32 * j);
                if (v4.x < best) { best = v4.x; bidx = mb; }
                if (v4.y < best) { best = v4.y; bidx = mb + 1; }
                if (v4.z < best) { best = v4.z; bidx = mb + 2; }
                if (v4.w < best) { best = v4.w; bidx = mb + 3; }
            }
#pragma unroll
            for (int off = 16; off > 0; off >>= 1) {
                float ov = __shfl_xor(best, off, 32);
                int   oi = __shfl_xor(bidx, off, 32);
                if (ov < best || (ov == best && oi < bidx)) { best = ov; bidx = oi; }
            }
            if (lane == 0) orow[kk] = bidx;
            if (((bidx >> 2) & 31) == lane) drow[bidx] = 3.4e38f;  // in-order DS per wave
        }
    }
}

// ---------------------------------------------------------------------------
// Kernel C: u'T[b][n][o] = ((W1-W2) @ p)[o][n] * inv[o] + (beta - mean*inv)[o]
//           v'T[b][n][o] = (W2 @ p)[o][n] * inv[o]
// Per wave: (mat, o-tile) fixed -> A fragment + BN constants hoisted.
// ---------------------------------------------------------------------------
__global__ __launch_bounds__(256) void gemm_kernel(const float* __restrict__ x,
                                                   const float* __restrict__ W,
                                                   const float* __restrict__ gamma,
                                                   const float* __restrict__ beta,
                                                   const float* __restrict__ rmean,
                                                   const float* __restrict__ rvar,
                                                   float* __restrict__ uT,
                                                   float* __restrict__ vT) {
    const int b    = blockIdx.y;
    const int nb0  = blockIdx.x * 128;
    const int tid  = threadIdx.x;
    const int wave = tid >> 5;
    const int lane = tid & 31;
    const int half = lane >> 4;
    const int l16  = lane & 15;

    const int mat = wave >> 2;                 // 0: u (W1-W2), 1: v (W2)
    const int ot  = wave & 3;                  // o tile
    const int o   = ot * 16 + l16;
    const float* xb = x + (size_t)b * CDIM * NPTS;

    v2f afrag[16];
#pragma unroll
    for (int ks = 0; ks < 16; ++ks) {
        int c0 = ks * 4 + half * 2;
        v2f a;
        if (mat == 0) {
            a.x = W[o * 128 + c0 + 0] - W[o * 128 + 64 + c0 + 0];
            a.y = W[o * 128 + c0 + 1] - W[o * 128 + 64 + c0 + 1];
        } else {
            a.x = W[o * 128 + 64 + c0 + 0];
            a.y = W[o * 128 + 64 + c0 + 1];
        }
        afrag[ks] = a;
    }
    float invs[8], adds[8];
#pragma unroll
    for (int i = 0; i < 8; ++i) {
        int oi = ot * 16 + i + 8 * half;
        float inv = gamma[oi] * __frsqrt_rn(rvar[oi] + BN_EPS);
        invs[i] = inv;
        adds[i] = (mat == 0) ? (beta[oi] - rmean[oi] * inv) : 0.0f;
    }
    float* base = (mat == 0 ? uT : vT) + (size_t)b * NPTS * ODIM;

    v2f buf[2][16];
#define LOAD_NT(bf, nt)                                           \
    {                                                             \
        int n_ = nb0 + (nt) * 16 + l16;                           \
        _Pragma("unroll") for (int ks = 0; ks < 16; ++ks) {       \
            int c0 = ks * 4 + half * 2;                           \
            (bf)[ks].x = xb[(c0 + 0) * NPTS + n_];                \
            (bf)[ks].y = xb[(c0 + 1) * NPTS + n_];                \
        }                                                         \
    }

    LOAD_NT(buf[0], 0)
#pragma unroll
    for (int nt = 0; nt < 8; ++nt) {
        const int cur = nt & 1, nxt = cur ^ 1;
        if (nt < 7) LOAD_NT(buf[nxt], nt + 1)
        v8f acc = {};
#pragma unroll
        for (int ks = 0; ks < 16; ++ks) acc = WMMA_F32(afrag[ks], buf[cur][ks], acc);
        const int n = nb0 + nt * 16 + l16;
        float* dst = base + (size_t)n * ODIM;
#pragma unroll
        for (int i = 0; i < 8; ++i) {
            int oi = ot * 16 + i + 8 * half;
            dst[oi] = acc[i] * invs[i] + adds[i];
        }
    }
#undef LOAD_NT
}

// ---------------------------------------------------------------------------
// Kernel D: out[b][o][n] = max_k leaky(u'T[b][n][o] + v'T[b][idx[b][n][k]][o])
// ---------------------------------------------------------------------------
__global__ __launch_bounds__(256) void fuse_kernel(const float* __restrict__ uT,
                                                   const float* __restrict__ vT,
                                                   const int* __restrict__ idx,
                                                   float* __restrict__ out) {
    __shared__ float tile[ODIM * 64];              // [o][j], 16 KB
    const int b    = blockIdx.y;
    const int n0   = blockIdx.x * 64;
    const int tid  = threadIdx.x;
    const int wave = tid >> 5;
    const int lane = tid & 31;

    const float2* uT2 = (const float2*)(uT + (size_t)b * NPTS * ODIM);
    const float2* vT2 = (const float2*)(vT + (size_t)b * NPTS * ODIM);

    for (int i = 0; i < 8; ++i) {
        int j = wave * 8 + i;                      // 0..63
        int n = n0 + j;
        const int* irow = idx + ((size_t)b * NPTS + n) * KNN;
        int mm[KNN];
#pragma unroll
        for (int kk = 0; kk < KNN; ++kk) mm[kk] = irow[kk];
        float2 vv[KNN];
#pragma unroll
        for (int kk = 0; kk < KNN; ++kk) vv[kk] = vT2[(size_t)mm[kk] * 32 + lane];
        float2 u = uT2[(size_t)n * 32 + lane];
        float m0v = -3.4e38f, m1v = -3.4e38f;
#pragma unroll
        for (int kk = 0; kk < KNN; ++kk) {
            float y0 = u.x + vv[kk].x;
            float y1 = u.y + vv[kk].y;
            y0 = (y0 >= 0.f) ? y0 : NEG_SLOPE * y0;
            y1 = (y1 >= 0.f) ? y1 : NEG_SLOPE * y1;
            m0v = fmaxf(m0v, y0);
            m1v = fmaxf(m1v, y1);
        }
        tile[(2 * lane) * 64 + j]     = m0v;
        tile[(2 * lane + 1) * 64 + j] = m1v;
    }
    __syncthreads();
#pragma unroll
    for (int r = 0; r < 16; ++r) {
        int e = r * 256 + tid;                     // 0..4095
        int o = e >> 6;
        int j = e & 63;
        out[((size_t)b * ODIM + o) * NPTS + n0 + j] = tile[o * 64 + j];
    }
}

// ---------------------------------------------------------------------------
extern "C" void kernel_launch(void* const* d_in, const int* in_sizes, int n_in,
                              void* d_out, int out_size, void* d_ws, size_t ws_size,
                              hipStream_t stream) {
    const float* x     = (const float*)d_in[0];   // (B, C, N)
    const float* W     = (const float*)d_in[1];   // (O, 2C)
    const float* gamma = (const float*)d_in[2];
    const float* beta  = (const float*)d_in[3];
    const float* rmean = (const float*)d_in[4];
    const float* rvar  = (const float*)d_in[5];
    (void)in_sizes; (void)n_in; (void)out_size; (void)ws_size;

    const int B = 8;
    char* ws = (char*)d_ws;
    float* sq  = (float*)ws;                                        // B*N
    int*   idx = (int*)(ws + (size_t)B * NPTS * 4);                 // B*N*20
    float* uT  = (float*)(ws + (size_t)B * NPTS * 4
                             + (size_t)B * NPTS * KNN * 4);         // B*N*64
    float* vT  = uT + (size_t)B * NPTS * ODIM;                      // B*N*64
    float* out = (float*)d_out;                                     // (B, O, N)

    sq_kernel<<<dim3(B * NPTS / 256), dim3(256), 0, stream>>>(x, sq);

    knn_kernel<<<dim3(NPTS / 16, B), dim3(512), 16 * NPTS * sizeof(float), stream>>>(
        x, sq, idx);

    gemm_kernel<<<dim3(NPTS / 128, B), dim3(256), 0, stream>>>(
        x, W, gamma, beta, rmean, rvar, uT, vT);

    fuse_kernel<<<dim3(NPTS / 64, B), dim3(256), 0, stream>>>(uT, vT, idx, out);
}